// SSIM3D_82351702934220
// MI455X (gfx1250) — compile-verified
//
#include <hip/hip_runtime.h>
#include <hip/hip_bf16.h>

typedef __attribute__((ext_vector_type(2))) float v2f;
typedef __attribute__((ext_vector_type(8))) float v8f;

#define N_DIM 2
#define D_DIM 160
#define H_DIM 192
#define W_DIM 160
#define HW    (H_DIM * W_DIM)                  // 30720
#define NE    (N_DIM * D_DIM * H_DIM * W_DIM)  // 9,830,400
#define COLS  (N_DIM * H_DIM * W_DIM)          // 61,440  (= 240 * 256 exactly)

// Normalized 11-tap Gaussian, sigma = 1.5 (matches reference _gauss_window)
__device__ __constant__ float GW[11] = {
    0.00102838f, 0.00759876f, 0.03600077f, 0.10936068f, 0.21300553f,
    0.26601172f, 0.21300553f, 0.10936068f, 0.03600077f, 0.00759876f,
    0.00102838f};

// Banded conv weight: Wband[j][n] = g[j-n] if 0 <= j-n <= 10 else 0
__device__ __forceinline__ float wband(int j, int n) {
  int k = j - n;
  int kc = ((unsigned)k <= 10u) ? k : 0;
  float g = GW[kc];
  return ((unsigned)k <= 10u) ? g : 0.0f;
}

// ---------------------------------------------------------------------------
// Kernel 1: per (n,d) slice, per 16x16 (h,w) tile: form 5 products, then
// W-axis and H-axis 11-tap Gaussian convs as banded WMMA f32 16x16x4 chains.
// One wave (32 lanes) per tile; EXEC all-ones around every WMMA.
// ---------------------------------------------------------------------------
__global__ __launch_bounds__(32)
void ssim3d_conv_wh(const float* __restrict__ img1,
                    const float* __restrict__ img2,
                    float* __restrict__ gbuf) {
  __shared__ float xs[5][32][32];  // 5 product tiles with halo
  __shared__ float ys[32][16];     // W-convolved intermediate (26 rows used)

  const int lane = threadIdx.x;
  const int w0 = blockIdx.x << 4;
  const int h0 = blockIdx.y << 4;
  const int z  = blockIdx.z;  // n*D + d
  const size_t plane = (size_t)z * HW;

  // Load 32x32 halo region (zero-padded) and form the 5 products.
  for (int r = 0; r < 32; ++r) {
    int h = h0 - 5 + r;
    int w = w0 - 5 + lane;
    float a = 0.0f, b = 0.0f;
    if (h >= 0 && h < H_DIM && w >= 0 && w < W_DIM) {
      size_t idx = plane + (size_t)h * W_DIM + (size_t)w;
      a = img1[idx];
      b = img2[idx];
    }
    xs[0][r][lane] = a;
    xs[1][r][lane] = b;
    xs[2][r][lane] = a * a;
    xs[3][r][lane] = b * b;
    xs[4][r][lane] = a * b;
  }
  __syncthreads();

  const int m    = lane & 15;  // row (A) / column (B,C,D) index
  const int half = lane >> 4;  // lane-half selects K pair / M+8

  // Banded-weight fragments; identical for the W-pass B operand and the
  // H-pass A operand since A_h[m][j] = W_w[j][m] = wband(j, m).
  v2f wfrag[8];
#pragma unroll
  for (int s = 0; s < 8; ++s) {
    int jb = 4 * s + 2 * half;
    wfrag[s][0] = wband(jb, m);
    wfrag[s][1] = wband(jb + 1, m);
  }

  for (int c = 0; c < 5; ++c) {
    // ---- W-axis conv: y(32x16) = X(32x32) * Wband(32x16), two row blocks
#pragma unroll
    for (int rb = 0; rb < 2; ++rb) {
      v8f acc = {};
#pragma unroll
      for (int s = 0; s < 8; ++s) {
        int jb = 4 * s + 2 * half;
        v2f av;
        av[0] = xs[c][rb * 16 + m][jb];
        av[1] = xs[c][rb * 16 + m][jb + 1];
        acc = __builtin_amdgcn_wmma_f32_16x16x4_f32(
            false, av, false, wfrag[s], (short)0, acc, false, false);
      }
#pragma unroll
      for (int v = 0; v < 8; ++v)
        ys[rb * 16 + v + 8 * half][m] = acc[v];
    }
    __syncthreads();

    // ---- H-axis conv: z(16x16) = Aband(16x32) * y(32x16)
    v8f acc2 = {};
#pragma unroll
    for (int s = 0; s < 8; ++s) {
      int jb = 4 * s + 2 * half;
      v2f bv;
      bv[0] = ys[jb][m];
      bv[1] = ys[jb + 1][m];
      acc2 = __builtin_amdgcn_wmma_f32_16x16x4_f32(
          false, wfrag[s], false, bv, (short)0, acc2, false, false);
    }

    float* dst = gbuf + (size_t)c * NE + plane;
#pragma unroll
    for (int v = 0; v < 8; ++v) {
      int hh = h0 + v + 8 * half;
      dst[(size_t)hh * W_DIM + (size_t)(w0 + m)] = acc2[v];
    }
    __syncthreads();
  }
}

// ---------------------------------------------------------------------------
// Kernel 2: D-axis conv (11-deep register ring buffer per channel) fused with
// the SSIM map and a block-level reduction. One thread per (n,h,w) column.
// The 5-channel intermediate (~197 MB) slightly exceeds the 192 MB L2, so
// prefetch the plane 16 steps ahead (speculative global_prefetch_b8) to hide
// the occasional HBM-latency miss; throttled to every 4th iteration since a
// cacheline covers all 32 lanes of a wave.
// ---------------------------------------------------------------------------
__global__ __launch_bounds__(256)
void ssim3d_conv_d(const float* __restrict__ gbuf,
                   float* __restrict__ partial) {
  __shared__ float red[256];
  const int tid = threadIdx.x;
  const int col = blockIdx.x * 256 + tid;  // COLS is an exact multiple of 256
  const int n   = col / HW;
  const int hw  = col - n * HW;
  const size_t base = (size_t)n * D_DIM * HW + (size_t)hw;

  float win[5][11];
#pragma unroll
  for (int c = 0; c < 5; ++c)
#pragma unroll
    for (int j = 0; j < 11; ++j)
      win[c][j] = 0.0f;

  // Invariant entering d: win[j] holds depth (d-6+j). Preload d = 0..4.
#pragma unroll
  for (int j = 6; j < 11; ++j) {
    size_t off = base + (size_t)(j - 6) * HW;
#pragma unroll
    for (int c = 0; c < 5; ++c)
      win[c][j] = gbuf[(size_t)c * NE + off];
  }

  float lsum = 0.0f;
  for (int d = 0; d < D_DIM; ++d) {
#pragma unroll
    for (int c = 0; c < 5; ++c)
#pragma unroll
      for (int j = 0; j < 10; ++j)
        win[c][j] = win[c][j + 1];
    int dn = d + 5;
    if (dn < D_DIM) {
      size_t off = base + (size_t)dn * HW;
#pragma unroll
      for (int c = 0; c < 5; ++c)
        win[c][10] = gbuf[(size_t)c * NE + off];
    } else {
#pragma unroll
      for (int c = 0; c < 5; ++c)
        win[c][10] = 0.0f;
    }

    // Software prefetch of the plane 16 steps ahead (d+5+16 < 160 guard).
    if ((d & 3) == 0 && (d + 21) < D_DIM) {
      size_t poff = base + (size_t)(d + 21) * HW;
#pragma unroll
      for (int c = 0; c < 5; ++c)
        __builtin_prefetch(&gbuf[(size_t)c * NE + poff], 0, 3);
    }

    float s[5];
#pragma unroll
    for (int c = 0; c < 5; ++c) {
      float acc = 0.0f;
#pragma unroll
      for (int j = 0; j < 11; ++j)
        acc = fmaf(GW[j], win[c][j], acc);
      s[c] = acc;
    }
    float mu1 = s[0], mu2 = s[1];
    float mu1s = mu1 * mu1, mu2s = mu2 * mu2, mu12 = mu1 * mu2;
    float sig1 = s[2] - mu1s, sig2 = s[3] - mu2s, sig12 = s[4] - mu12;
    const float C1v = 1.0e-4f, C2v = 9.0e-4f;
    float num = (2.0f * mu12 + C1v) * (2.0f * sig12 + C2v);
    float den = (mu1s + mu2s + C1v) * (sig1 + sig2 + C2v);
    lsum += num / den;
  }

  red[tid] = lsum;
  __syncthreads();
#pragma unroll
  for (int stn = 128; stn > 0; stn >>= 1) {
    if (tid < stn) red[tid] += red[tid + stn];
    __syncthreads();
  }
  if (tid == 0) partial[blockIdx.x] = red[0];
}

// ---------------------------------------------------------------------------
// Kernel 3: final reduction of 240 partials -> mean
// ---------------------------------------------------------------------------
__global__ __launch_bounds__(256)
void ssim3d_reduce(const float* __restrict__ partial,
                   float* __restrict__ out) {
  __shared__ float red[256];
  int tid = threadIdx.x;
  red[tid] = (tid < (COLS / 256)) ? partial[tid] : 0.0f;
  __syncthreads();
#pragma unroll
  for (int stn = 128; stn > 0; stn >>= 1) {
    if (tid < stn) red[tid] += red[tid + stn];
    __syncthreads();
  }
  if (tid == 0) out[0] = red[0] * (1.0f / (float)NE);
}

extern "C" void kernel_launch(void* const* d_in, const int* in_sizes, int n_in,
                              void* d_out, int out_size, void* d_ws,
                              size_t ws_size, hipStream_t stream) {
  const float* img1 = (const float*)d_in[0];
  const float* img2 = (const float*)d_in[1];
  float* out = (float*)d_out;
  float* gbuf = (float*)d_ws;                     // 5 * NE floats (~197 MB)
  float* partial = gbuf + (size_t)5 * NE;         // 240 floats

  dim3 g1(W_DIM / 16, H_DIM / 16, N_DIM * D_DIM); // (10, 12, 320)
  ssim3d_conv_wh<<<g1, 32, 0, stream>>>(img1, img2, gbuf);
  ssim3d_conv_d<<<COLS / 256, 256, 0, stream>>>(gbuf, partial);
  ssim3d_reduce<<<1, 256, 0, stream>>>(partial, out);
}